// DCWTv2Attention_36464272343168
// MI455X (gfx1250) — compile-verified
//
#include <hip/hip_runtime.h>
#include <hip/hip_bf16.h>
#include <math.h>

// ---------------- problem constants (match reference) ----------------
#define B_      2
#define L_      2048
#define DM_     512
#define H_      8
#define D_      64
#define KMAX_   2
#define E_      32
#define LOGN_   12
#define TREE_   8192
#define LEAF_   4096
#define NDEPTH_ 13
#define CMAX_   12          // true max cover is 11; +1 slack
#define PI_     3.14159265358979323846f

typedef float v2f __attribute__((ext_vector_type(2)));
typedef float v8f __attribute__((ext_vector_type(8)));
typedef int   v4i __attribute__((ext_vector_type(4)));

// pointer-to-vector types in explicit address spaces for the async builtin
typedef __attribute__((address_space(1))) v4i* as1_v4i_ptr;
typedef __attribute__((address_space(3))) v4i* as3_v4i_ptr;

// ---------------------------------------------------------------------
// CDNA5 async global->LDS copy helpers (ASYNCcnt-tracked).  Falls back
// to synchronous 16B copies if the builtins are not declared.
// ---------------------------------------------------------------------
#if __has_builtin(__builtin_amdgcn_global_load_async_to_lds_b128)
#define HAVE_ASYNC_LDS 1
#else
#define HAVE_ASYNC_LDS 0
#endif

__device__ __forceinline__ void lds_copy16(const float* g, float* l) {
#if HAVE_ASYNC_LDS
  __builtin_amdgcn_global_load_async_to_lds_b128(
      (as1_v4i_ptr)g, (as3_v4i_ptr)l, /*offset=*/0, /*cpol=*/0);
#else
  *(float4*)l = *(const float4*)g;
#endif
}

__device__ __forceinline__ void lds_copy_wait() {
#if HAVE_ASYNC_LDS
#if __has_builtin(__builtin_amdgcn_s_wait_asynccnt)
  __builtin_amdgcn_s_wait_asynccnt(0);
#else
  asm volatile("s_wait_asynccnt 0x0" ::: "memory");
#endif
#endif
}

// ---------------------------------------------------------------------
// Zero the tree buffer (harness poisons ws with 0xAA).
// ---------------------------------------------------------------------
__global__ void zero_tree_kernel(float* __restrict__ p, int n) {
  int i = blockIdx.x * blockDim.x + threadIdx.x;
  int stride = gridDim.x * blockDim.x;
  for (; i < n; i += stride) p[i] = 0.0f;
}

// ---------------------------------------------------------------------
// Build cover tables (segment-tree cover of [LEAF, LEAF+t+1)) and the
// sinusoidal depth embedding (module-level constants in the reference).
// ---------------------------------------------------------------------
__global__ void build_tables_kernel(int* __restrict__ cnodes,
                                    int* __restrict__ cdeps,
                                    int* __restrict__ ccnt,
                                    float* __restrict__ dE) {
  int t = blockIdx.x * blockDim.x + threadIdx.x;
  if (t < L_) {
    int l = LEAF_, r = LEAF_ + t + 1;
    int c = 0;
    while (l < r) {
      if (l & 1) { if (c < CMAX_) { cnodes[t * CMAX_ + c] = l; c++; } l++; }
      if (r & 1) { r--; if (c < CMAX_) { cnodes[t * CMAX_ + c] = r; c++; } }
      l >>= 1; r >>= 1;
    }
    ccnt[t] = c;
    for (int i = 0; i < c; ++i) {
      int node = cnodes[t * CMAX_ + i];
      int lg = 31 - __clz(node);
      cdeps[t * CMAX_ + i] = LOGN_ - lg;
    }
    for (int i = c; i < CMAX_; ++i) { cnodes[t * CMAX_ + i] = LEAF_; cdeps[t * CMAX_ + i] = 0; }
  }
  if (t < NDEPTH_ * E_) {
    int n = t / E_, j = t % E_;
    float inv = expf(-((float)(2 * (j / 2))) * (logf(10000.0f) / (float)E_));
    float arg = (float)n * inv;
    dE[t] = (j & 1) ? cosf(arg) : sinf(arg);
  }
}

// ---------------------------------------------------------------------
// C[M,N] = A[M,K] @ W[N,K]^T, fp32 exact via V_WMMA_F32_16X16X4_F32.
//
// 256 threads = 8 waves; block tile 32x64 (wave w -> 16x16 tile
// (w/4, w%4)).  K is consumed in 64-wide slabs that are double-buffered
// in LDS and filled with ASYNCcnt-tracked b128 async loads; compute on
// slab s overlaps the async fill of slab s+1, then s_wait_asynccnt 0 +
// barrier.  Rows padded to 68 floats (272B): keeps every b128 LDS write
// 16B-aligned and gives the 16-lane fragment reads stride-4 banks
// (conflict-free).
// ---------------------------------------------------------------------
#define TM_ 32
#define TN_ 64
#define TK_ 64
#define PAD_ 68

__global__ __launch_bounds__(256)
void wmma_gemm_xwT_kernel(const float* __restrict__ A,
                          const float* __restrict__ W,
                          float* __restrict__ C,
                          int M, int N, int K) {
  __shared__ __align__(16) float sA[2][TM_][PAD_];
  __shared__ __align__(16) float sB[2][TN_][PAD_];

  int tid  = threadIdx.x;
  int wave = tid >> 5;
  int lane = tid & 31;
  int half = lane >> 4;    // selects K pair inside the fragment
  int l16  = lane & 15;
  int mb = blockIdx.x * TM_;
  int nb = blockIdx.y * TN_;
  int wm = (wave >> 2) * 16;
  int wn = (wave & 3) * 16;

  auto loadStep = [&](int buf, int k0) {
    // A slab: 32x64 floats = 512 b128 chunks; 2 per thread
#pragma unroll
    for (int i = 0; i < 2; ++i) {
      int c = tid + 256 * i;
      int r = c >> 4, col = (c & 15) * 4;
      lds_copy16(A + (size_t)(mb + r) * K + k0 + col, &sA[buf][r][col]);
    }
    // B slab: 64x64 floats = 1024 b128 chunks; 4 per thread
#pragma unroll
    for (int i = 0; i < 4; ++i) {
      int c = tid + 256 * i;
      int r = c >> 4, col = (c & 15) * 4;
      lds_copy16(W + (size_t)(nb + r) * K + k0 + col, &sB[buf][r][col]);
    }
  };

  v8f acc = {};
  loadStep(0, 0);
  lds_copy_wait();
  __syncthreads();

  int steps = K / TK_;
  for (int s = 0; s < steps; ++s) {
    int buf = s & 1;
    if (s + 1 < steps) loadStep(buf ^ 1, (s + 1) * TK_);

    const float* aBase = &sA[buf][wm + l16][half * 2];
    const float* bBase = &sB[buf][wn + l16][half * 2];
#pragma unroll
    for (int kk = 0; kk < TK_; kk += 4) {
      v2f a = *(const v2f*)(aBase + kk);   // ds_load_b64
      v2f b = *(const v2f*)(bBase + kk);   // ds_load_b64
      acc = __builtin_amdgcn_wmma_f32_16x16x4_f32(
          /*neg_a=*/false, a, /*neg_b=*/false, b,
          /*c_mod=*/(short)0, acc, /*reuse_a=*/false, /*reuse_b=*/false);
    }
    lds_copy_wait();
    __syncthreads();
  }

  // C/D layout: lane%16 = N, VGPR r -> M = r + (lane/16)*8
  float* cBase = C + (size_t)(mb + wm + half * 8) * N + nb + wn + l16;
#pragma unroll
  for (int r = 0; r < 8; ++r) cBase[(size_t)r * N] = acc[r];
}

// ---------------------------------------------------------------------
// Scatter v (B*L, 512) into tree leaves: tree[LEAF+l, b, h, 0, :] = v
// ---------------------------------------------------------------------
__global__ void scatter_leaves_kernel(const float* __restrict__ v,
                                      float* __restrict__ tree) {
  int i = blockIdx.x * blockDim.x + threadIdx.x;
  if (i >= B_ * L_ * DM_) return;
  int col = i % DM_, row = i / DM_;
  int d = col % D_, h = col / D_;
  int l = row % L_, b = row / L_;
  size_t off = ((((size_t)(LEAF_ + l)) * B_ + b) * H_ + h) * (KMAX_ * D_) + d;
  tree[off] = v[i];
}

// ---------------------------------------------------------------------
// One tree-merge level.  Block = 64 threads (2 wave32), one block per
// (parent node, b, h).  All small matvecs staged through LDS.
// ---------------------------------------------------------------------
__global__ __launch_bounds__(64)
void merge_kernel(float* __restrict__ tree,
                  const float* __restrict__ glw, const float* __restrict__ glb,
                  const float* __restrict__ grw, const float* __restrict__ grb,
                  const float* __restrict__ pqb,
                  const float* __restrict__ qow, const float* __restrict__ qob,
                  const float* __restrict__ lng, const float* __restrict__ lnb,
                  const float* __restrict__ skw, const float* __restrict__ skalpha,
                  const float* __restrict__ wf, const float* __restrict__ wd,
                  const float* __restrict__ wp, const float* __restrict__ dE,
                  int depth, int lo, int kc) {
  __shared__ float s_gin[2 * D_ + E_];     // [lm | rm | depth-embed]
  __shared__ float s_bank[2 * KMAX_][D_];
  __shared__ float s_pq[KMAX_][D_];
  __shared__ float s_logit[KMAX_][2 * KMAX_];
  __shared__ float s_attn[KMAX_][2 * KMAX_];
  __shared__ float s_raw[KMAX_][D_];

  int j = threadIdx.x;                     // 0..63 : the D dimension
  int bh = blockIdx.y;
  int b = bh / H_, h = bh % H_;
  int node = lo + blockIdx.x;

  size_t nstride = (size_t)KMAX_ * D_;
  const float* Lc = tree + ((((size_t)(2 * node)) * B_ + b) * H_ + h) * nstride;
  const float* Rc = tree + ((((size_t)(2 * node + 1)) * B_ + b) * H_ + h) * nstride;

  // wave rotation (scale cancels: alpha*scale = softplus, omega*scale = freq)
  float sp = log1pf(expf(wd[h]));
  float decay = expf(-sp);
  float ang = wf[h] + wp[h] + (float)depth * (PI_ * 0.25f);
  float pr = decay * cosf(ang);
  float pi = decay * sinf(ang);

  float fl[KMAX_], rot[KMAX_];
  float lm = 0.f, rm = 0.f;
#pragma unroll
  for (int k = 0; k < KMAX_; ++k) {
    if (k < kc) {
      fl[k] = Lc[k * D_ + j];
      float own = Rc[k * D_ + j];
      float par = Rc[k * D_ + (j ^ 32)];
      rot[k] = (j < 32) ? (pr * own - pi * par) : (pi * par + pr * own);
      lm += fl[k]; rm += rot[k];
    } else { fl[k] = 0.f; rot[k] = 0.f; }
  }
  float inv_kc = 1.0f / (float)kc;
  lm *= inv_kc; rm *= inv_kc;

  s_gin[j] = lm;
  s_gin[D_ + j] = rm;
  if (j < E_) s_gin[2 * D_ + j] = dE[depth * E_ + j];
  __syncthreads();

  // gates: sigmoid(gate_in @ W.T + b), W is (64,160) row-major
  float al = glb[j], ar = grb[j];
#pragma unroll 4
  for (int i = 0; i < 2 * D_ + E_; ++i) {
    float gi = s_gin[i];
    al += gi * glw[j * (2 * D_ + E_) + i];
    ar += gi * grw[j * (2 * D_ + E_) + i];
  }
  float g_l = 1.0f / (1.0f + expf(-al));
  float g_r = 1.0f / (1.0f + expf(-ar));

  for (int r = 0; r < kc; ++r) {
    s_bank[r][j]      = fl[r] * g_l;
    s_bank[kc + r][j] = rot[r] * g_r;
  }

  // parent queries: pq[q] = base[q] + query_offset_w @ dv + b
  float pq0 = qob[j], pq1 = qob[D_ + j];
#pragma unroll 4
  for (int e = 0; e < E_; ++e) {
    float dv = s_gin[2 * D_ + e];
    pq0 += qow[j * E_ + e] * dv;
    pq1 += qow[(D_ + j) * E_ + e] * dv;
  }
  s_pq[0][j] = pqb[j] + pq0;
  s_pq[1][j] = pqb[D_ + j] + pq1;
  __syncthreads();

  int nb = 2 * kc;
  if (j < KMAX_ * nb) {
    int q = j / nb, r = j % nb;
    float acc = 0.f;
#pragma unroll 8
    for (int i = 0; i < D_; ++i) acc += s_pq[q][i] * s_bank[r][i];
    s_logit[q][r] = acc * 0.125f;          // 1/sqrt(64)
  }
  __syncthreads();

  if (j < KMAX_) {
    float m = -1e30f;
    for (int r = 0; r < nb; ++r) m = fmaxf(m, s_logit[j][r]);
    float s = 0.f;
    for (int r = 0; r < nb; ++r) { float e = expf(s_logit[j][r] - m); s_attn[j][r] = e; s += e; }
    float inv = 1.0f / s;
    for (int r = 0; r < nb; ++r) s_attn[j][r] *= inv;
  }
  __syncthreads();

  float raw0 = 0.f, raw1 = 0.f;
  for (int r = 0; r < nb; ++r) {
    float bv = s_bank[r][j];
    raw0 += s_attn[0][r] * bv;
    raw1 += s_attn[1][r] * bv;
  }
  s_raw[0][j] = raw0; s_raw[1][j] = raw1;
  __syncthreads();

  float skp = 0.f;
#pragma unroll 8
  for (int i = 0; i < D_; ++i) skp += s_gin[i] * skw[j * D_ + i];
  float sa = 1.0f / (1.0f + expf(-skalpha[0]));

  float* Pn = tree + (((size_t)node * B_ + b) * H_ + h) * nstride;
#pragma unroll
  for (int q = 0; q < KMAX_; ++q) {
    float sum = 0.f, sq = 0.f;
    for (int i = 0; i < D_; ++i) { float x = s_raw[q][i]; sum += x; sq += x * x; }
    float mu = sum * (1.0f / D_);
    float var = sq * (1.0f / D_) - mu * mu;
    float x = (q == 0) ? raw0 : raw1;
    float normed = (x - mu) * rsqrtf(var + 1e-5f) * lng[j] + lnb[j];
    Pn[q * D_ + j] = normed + sa * skp;
  }
}

// ---------------------------------------------------------------------
// Cover attention.  Block = 256 threads = 8 waves; one wave per
// (b,l,h).  Depth-projected queries computed on the fly (q + P_dep q),
// logits reduced with wave32 xor-shuffle butterflies, softmax over the
// C*K_MAX masked logit set, context written to ctx (B*L, 512).
// ---------------------------------------------------------------------
__global__ __launch_bounds__(256)
void cover_attn_kernel(const float* __restrict__ tree,
                       const float* __restrict__ qbuf,
                       const int* __restrict__ cnodes,
                       const int* __restrict__ cdeps,
                       const int* __restrict__ ccnt,
                       const float* __restrict__ Pw,      // (13,64,64)
                       const float* __restrict__ dtemp,   // (13)
                       float* __restrict__ ctx) {
  __shared__ float sq[8][D_];

  int wave = threadIdx.x >> 5;
  int t    = threadIdx.x & 31;
  int idx  = blockIdx.x * 8 + wave;        // (b,l,h) flat
  int h = idx % H_;
  int l = (idx / H_) % L_;
  int b = idx / (H_ * L_);

  size_t qrow = ((size_t)b * L_ + l) * DM_ + h * D_;
  sq[wave][t]      = qbuf[qrow + t];
  sq[wave][t + 32] = qbuf[qrow + t + 32];
  __syncthreads();

  int cnt = ccnt[l];
  float logit[CMAX_ * KMAX_];
  float vv0[CMAX_ * KMAX_], vv1[CMAX_ * KMAX_];

  for (int c = 0; c < CMAX_; ++c) {
    int node = cnodes[l * CMAX_ + c];
    int dep  = cdeps[l * CMAX_ + c];
    int kval = (c < cnt) ? ((dep == 0) ? 1 : 2) : 0;

    // q_all[o] = q[o] + sum_d q[d] * P[dep][o][d]   for o = t, t+32
    const float* P = Pw + (size_t)dep * D_ * D_;
    float qa0 = 0.f, qa1 = 0.f;
#pragma unroll 8
    for (int dd = 0; dd < D_; ++dd) {
      float qv = sq[wave][dd];
      qa0 += qv * P[t * D_ + dd];
      qa1 += qv * P[(t + 32) * D_ + dd];
    }
    qa0 += sq[wave][t];
    qa1 += sq[wave][t + 32];

    float sc = 1.0f / ((log1pf(expf(dtemp[dep])) + 1e-6f) * 8.0f);

    const float* V = tree + (((size_t)node * B_ + b) * H_ + h) * (KMAX_ * D_);
#pragma unroll
    for (int k = 0; k < KMAX_; ++k) {
      float v0 = V[k * D_ + t];
      float v1 = V[k * D_ + t + 32];
      vv0[c * KMAX_ + k] = v0;
      vv1[c * KMAX_ + k] = v1;
      float p = qa0 * v0 + qa1 * v1;
#pragma unroll
      for (int off = 16; off > 0; off >>= 1) p += __shfl_xor(p, off, 32);
      logit[c * KMAX_ + k] = (k < kval) ? p * sc : -1e9f;
    }
  }

  float m = -1e30f;
#pragma unroll
  for (int i = 0; i < CMAX_ * KMAX_; ++i) m = fmaxf(m, logit[i]);
  float s = 0.f;
#pragma unroll
  for (int i = 0; i < CMAX_ * KMAX_; ++i) { float e = expf(logit[i] - m); logit[i] = e; s += e; }
  float inv = 1.0f / s;

  float c0 = 0.f, c1 = 0.f;
#pragma unroll
  for (int i = 0; i < CMAX_ * KMAX_; ++i) {
    float w = logit[i] * inv;
    c0 += w * vv0[i];
    c1 += w * vv1[i];
  }

  size_t orow = ((size_t)b * L_ + l) * DM_ + h * D_;
  ctx[orow + t]      = c0;
  ctx[orow + t + 32] = c1;
}

// ---------------------------------------------------------------------
// Host launcher
// ---------------------------------------------------------------------
extern "C" void kernel_launch(void* const* d_in, const int* in_sizes, int n_in,
                              void* d_out, int out_size, void* d_ws, size_t ws_size,
                              hipStream_t stream) {
  const float* x    = (const float*)d_in[0];
  const float* Wq   = (const float*)d_in[1];
  const float* Wv   = (const float*)d_in[2];
  const float* Wo   = (const float*)d_in[3];
  const float* glw  = (const float*)d_in[4];
  const float* glb  = (const float*)d_in[5];
  const float* grw  = (const float*)d_in[6];
  const float* grb  = (const float*)d_in[7];
  const float* pqb  = (const float*)d_in[8];
  const float* qow  = (const float*)d_in[9];
  const float* qob  = (const float*)d_in[10];
  const float* lng  = (const float*)d_in[11];
  const float* lnb  = (const float*)d_in[12];
  const float* skw  = (const float*)d_in[13];
  const float* ska  = (const float*)d_in[14];
  const float* wf   = (const float*)d_in[15];
  const float* wd   = (const float*)d_in[16];
  const float* wp   = (const float*)d_in[17];
  const float* Pw   = (const float*)d_in[18];
  const float* dtmp = (const float*)d_in[19];
  float* out = (float*)d_out;

  // ------------- workspace layout (floats) -------------
  const size_t TREE_ELEMS = (size_t)TREE_ * B_ * H_ * KMAX_ * D_;   // 16,777,216
  const size_t MAT_ELEMS  = (size_t)B_ * L_ * DM_;                  //  2,097,152
  float* tree = (float*)d_ws;
  float* vbuf = tree + TREE_ELEMS;
  float* qbuf = vbuf + MAT_ELEMS;
  float* ctx  = qbuf + MAT_ELEMS;
  float* dE   = ctx + MAT_ELEMS;
  int* cnodes = (int*)(dE + NDEPTH_ * E_);
  int* cdeps  = cnodes + L_ * CMAX_;
  int* ccnt   = cdeps + L_ * CMAX_;

  // 1) zero the tree (poisoned workspace)
  zero_tree_kernel<<<4096, 256, 0, stream>>>(tree, (int)TREE_ELEMS);

  // 2) cover tables + depth embedding
  build_tables_kernel<<<(L_ + 255) / 256, 256, 0, stream>>>(cnodes, cdeps, ccnt, dE);

  // 3) v = x @ Wv.T ; q = x @ Wq.T   (fp32 WMMA GEMMs, async-LDS staged)
  dim3 gemmGrid(B_ * L_ / TM_, DM_ / TN_);
  wmma_gemm_xwT_kernel<<<gemmGrid, 256, 0, stream>>>(x, Wv, vbuf, B_ * L_, DM_, DM_);
  wmma_gemm_xwT_kernel<<<gemmGrid, 256, 0, stream>>>(x, Wq, qbuf, B_ * L_, DM_, DM_);

  // 4) scatter v into tree leaves
  scatter_leaves_kernel<<<(B_ * L_ * DM_ + 255) / 256, 256, 0, stream>>>(vbuf, tree);

  // 5) bottom-up merge, levels d = 1..LOG_N
  for (int d = 1; d <= LOGN_; ++d) {
    int lo = 1 << (LOGN_ - d);
    int kc = (d == 1) ? 1 : 2;
    merge_kernel<<<dim3(lo, B_ * H_), 64, 0, stream>>>(
        tree, glw, glb, grw, grb, pqb, qow, qob, lng, lnb, skw, ska,
        wf, wd, wp, dE, d, lo, kc);
  }

  // 6) cover attention -> ctx
  cover_attn_kernel<<<(B_ * L_ * H_) / 8, 256, 0, stream>>>(
      tree, qbuf, cnodes, cdeps, ccnt, Pw, dtmp, ctx);

  // 7) out = ctx @ Wo.T
  wmma_gemm_xwT_kernel<<<gemmGrid, 256, 0, stream>>>(ctx, Wo, out, B_ * L_, DM_, DM_);
}